// ZBL_repulsion_21861383536753
// MI455X (gfx1250) — compile-verified
//
#include <hip/hip_runtime.h>

// ZBL repulsion: per-pair screened-Coulomb energy + segment-sum over sorted idx_i.
// Roofline: 3.2M pairs * 16B streamed = 51.2 MB @ 23.3 TB/s => ~2.2 us floor.
// ~128 MFLOP + 12.8M v_exp_f32 total => purely HBM-bound; WMMA inapplicable.
// CDNA5 specialization: double-buffered GLOBAL_LOAD_ASYNC_TO_LDS_B128 staging
// (ASYNCcnt-tracked, no VGPR landing slots), ds_load_b128 readback, L2-resident
// float2 {Z, Z^|a|} gather table, run-length-compacted global_atomic_add_f32.

#define ZBL_BLOCK 256
#define ZBL_CHUNK 8                       // pairs per thread
#define ZBL_TILE  (ZBL_BLOCK * ZBL_CHUNK) // 2048 pairs per block tile
#define ZBL_MAIN_GRID 768                 // persistent blocks for the async pipeline

// ---------------------------------------------------------------------------
// Shared scalar parameters
// ---------------------------------------------------------------------------
struct ZblConsts {
    float inv_ac;
    float c0, c1, c2, c3;
    float e0, e1, e2, e3;
    float pref;
};

__device__ __forceinline__ ZblConsts zbl_consts(const float* a_coef,
                                                const float* phi_c,
                                                const float* phi_e,
                                                const float* ke,
                                                const float* d2a,
                                                const float* e2m) {
    ZblConsts K;
    K.inv_ac = 1.0f / fabsf(a_coef[0]);
    float c0 = fabsf(phi_c[0]), c1 = fabsf(phi_c[1]);
    float c2 = fabsf(phi_c[2]), c3 = fabsf(phi_c[3]);
    float inv_cs = 1.0f / (c0 + c1 + c2 + c3);
    K.c0 = c0 * inv_cs; K.c1 = c1 * inv_cs;
    K.c2 = c2 * inv_cs; K.c3 = c3 * inv_cs;
    K.e0 = fabsf(phi_e[0]); K.e1 = fabsf(phi_e[1]);
    K.e2 = fabsf(phi_e[2]); K.e3 = fabsf(phi_e[3]);
    // repulsion = ke * Zi * Zj / (d * d2a) * phi * cut, then scaled by e2m
    K.pref = ke[0] * e2m[0] / d2a[0];
    return K;
}

__device__ __forceinline__ float zbl_pair_val(float d, float cu, int ii, int jj,
                                              const float2* __restrict__ zza,
                                              const ZblConsts& K) {
    float2 zi = zza[ii];                  // L2-hit gathers (table is 800 KB)
    float2 zj = zza[jj];
    float arg = d * (zi.y + zj.y) * K.inv_ac;
    float phi = K.c0 * __expf(-K.e0 * arg) + K.c1 * __expf(-K.e1 * arg)
              + K.c2 * __expf(-K.e2 * arg) + K.c3 * __expf(-K.e3 * arg);
    // v_rcp_f32 (~1 ulp) instead of the ~10-op precise-divide expansion
    return K.pref * zi.x * zj.x * phi * cu * __builtin_amdgcn_rcpf(d);
}

// ---------------------------------------------------------------------------
// CDNA5 async global -> LDS staging (GVS mode: s[base] + 32-bit vgpr offset)
// ---------------------------------------------------------------------------
__device__ __forceinline__ void zbl_async_b128(unsigned lds_byte, unsigned gbyte,
                                               const void* base) {
    asm volatile("global_load_async_to_lds_b128 %0, %1, %2"
                 :: "v"(lds_byte), "v"(gbyte), "s"((unsigned long long)base)
                 : "memory");
}

// ---------------------------------------------------------------------------
// Kernel 1: per-atom table {Z, Z^|a_exp|} (L2-resident) + zero the output.
// ---------------------------------------------------------------------------
__global__ __launch_bounds__(ZBL_BLOCK) void zbl_prep(
    const int*   __restrict__ Z,
    const float* __restrict__ a_exp,
    float2*      __restrict__ zza,
    float*       __restrict__ out,
    int n_atoms)
{
    int i = blockIdx.x * ZBL_BLOCK + threadIdx.x;
    if (i >= n_atoms) return;
    float e  = fabsf(a_exp[0]);
    float z  = (float)Z[i];
    float za = powf(z, e);                // one-time cost, full precision
    zza[i]   = make_float2(z, za);
    out[i]   = 0.0f;
}

// ---------------------------------------------------------------------------
// Kernel 2: main pair stream. Persistent blocks grid-stride over full tiles
// with double-buffered async LDS staging. Each thread stages and consumes only
// its own 8-pair slice -> per-wave s_wait_asynccnt is the only sync needed
// (async loads complete in order within a wave).
// ---------------------------------------------------------------------------
__global__ __launch_bounds__(ZBL_BLOCK) void zbl_pairs_main(
    const float*  __restrict__ dist,
    const float*  __restrict__ cut,
    const int*    __restrict__ idx_i,
    const int*    __restrict__ idx_j,
    const float2* __restrict__ zza,
    const float*  __restrict__ a_coef,
    const float*  __restrict__ phi_c,
    const float*  __restrict__ phi_e,
    const float*  __restrict__ ke,
    const float*  __restrict__ d2a,
    const float*  __restrict__ e2m,
    float*        __restrict__ out,
    int ntiles)
{
    __shared__ __align__(16) float sD[2][ZBL_TILE];
    __shared__ __align__(16) float sC[2][ZBL_TILE];
    __shared__ __align__(16) int   sI[2][ZBL_TILE];
    __shared__ __align__(16) int   sJ[2][ZBL_TILE];

    const ZblConsts K = zbl_consts(a_coef, phi_c, phi_e, ke, d2a, e2m);
    const int tid  = threadIdx.x;
    const int slot = tid * ZBL_CHUNK;     // this thread's tile-local pair base

    // Stage one tile's worth of this thread's slice: 2 x b128 per array.
    auto issue_tile = [&](int t, int b) {
        unsigned g = ((unsigned)t * (unsigned)ZBL_TILE + (unsigned)slot) * 4u;
        zbl_async_b128((unsigned)(unsigned long long)&sD[b][slot],     g,      dist);
        zbl_async_b128((unsigned)(unsigned long long)&sD[b][slot + 4], g + 16, dist);
        zbl_async_b128((unsigned)(unsigned long long)&sC[b][slot],     g,      cut);
        zbl_async_b128((unsigned)(unsigned long long)&sC[b][slot + 4], g + 16, cut);
        zbl_async_b128((unsigned)(unsigned long long)&sI[b][slot],     g,      idx_i);
        zbl_async_b128((unsigned)(unsigned long long)&sI[b][slot + 4], g + 16, idx_i);
        zbl_async_b128((unsigned)(unsigned long long)&sJ[b][slot],     g,      idx_j);
        zbl_async_b128((unsigned)(unsigned long long)&sJ[b][slot + 4], g + 16, idx_j);
    };

    int tile = blockIdx.x;
    if (tile >= ntiles) return;
    int buf = 0;
    issue_tile(tile, 0);                  // prime the pipeline (8 async ops)

    for (; tile < ntiles; tile += (int)gridDim.x) {
        int next = tile + (int)gridDim.x;
        if (next < ntiles) {
            issue_tile(next, buf ^ 1);    // prefetch tile t+1 into other buffer
            // 16 outstanding; current tile's 8 complete first (in-order).
            asm volatile("s_wait_asynccnt 0x8" ::: "memory");
        } else {
            asm volatile("s_wait_asynccnt 0x0" ::: "memory");
        }

        // ds_load_b128 readback of this thread's slice.
        float dv[ZBL_CHUNK], cv[ZBL_CHUNK];
        int   iv[ZBL_CHUNK], jv[ZBL_CHUNK];
        *(float4*)(dv)     = *(const float4*)&sD[buf][slot];
        *(float4*)(dv + 4) = *(const float4*)&sD[buf][slot + 4];
        *(float4*)(cv)     = *(const float4*)&sC[buf][slot];
        *(float4*)(cv + 4) = *(const float4*)&sC[buf][slot + 4];
        *(int4*)(iv)       = *(const int4*)&sI[buf][slot];
        *(int4*)(iv + 4)   = *(const int4*)&sI[buf][slot + 4];
        *(int4*)(jv)       = *(const int4*)&sJ[buf][slot];
        *(int4*)(jv + 4)   = *(const int4*)&sJ[buf][slot + 4];

        // Run-length compaction over sorted idx_i before global atomics.
        int   cur = iv[0];
        float acc = 0.0f;
        #pragma unroll
        for (int k = 0; k < ZBL_CHUNK; ++k) {
            float v = zbl_pair_val(dv[k], cv[k], iv[k], jv[k], zza, K);
            if (iv[k] != cur) {
                atomicAdd(out + cur, acc);  // global_atomic_add_f32 (no return)
                cur = iv[k];
                acc = v;
            } else {
                acc += v;
            }
        }
        atomicAdd(out + cur, acc);
        buf ^= 1;
    }
}

// ---------------------------------------------------------------------------
// Kernel 3: tail pairs [base, n_pairs), < ZBL_TILE of them: plain scalar path
// so the async path never reads out of bounds.
// ---------------------------------------------------------------------------
__global__ __launch_bounds__(ZBL_BLOCK) void zbl_pairs_tail(
    const float*  __restrict__ dist,
    const float*  __restrict__ cut,
    const int*    __restrict__ idx_i,
    const int*    __restrict__ idx_j,
    const float2* __restrict__ zza,
    const float*  __restrict__ a_coef,
    const float*  __restrict__ phi_c,
    const float*  __restrict__ phi_e,
    const float*  __restrict__ ke,
    const float*  __restrict__ d2a,
    const float*  __restrict__ e2m,
    float*        __restrict__ out,
    long long base, long long n_pairs)
{
    const ZblConsts K = zbl_consts(a_coef, phi_c, phi_e, ke, d2a, e2m);
    long long p0 = base + (long long)threadIdx.x * ZBL_CHUNK;
    if (p0 >= n_pairs) return;
    int   cur = idx_i[p0];
    float acc = 0.0f;
    for (int k = 0; k < ZBL_CHUNK; ++k) {
        long long p = p0 + k;
        if (p >= n_pairs) break;
        int ii = idx_i[p];
        float v = zbl_pair_val(dist[p], cut[p], ii, idx_j[p], zza, K);
        if (ii != cur) {
            atomicAdd(out + cur, acc);
            cur = ii;
            acc = v;
        } else {
            acc += v;
        }
    }
    atomicAdd(out + cur, acc);
}

// ---------------------------------------------------------------------------
extern "C" void kernel_launch(void* const* d_in, const int* in_sizes, int n_in,
                              void* d_out, int out_size, void* d_ws, size_t ws_size,
                              hipStream_t stream) {
    (void)n_in; (void)out_size; (void)ws_size;

    const int*   Z      = (const int*)  d_in[0];
    const float* dist   = (const float*)d_in[1];
    const float* cut    = (const float*)d_in[2];
    const int*   idx_i  = (const int*)  d_in[3];
    const int*   idx_j  = (const int*)  d_in[4];
    const float* a_coef = (const float*)d_in[5];
    const float* a_exp  = (const float*)d_in[6];
    const float* phi_c  = (const float*)d_in[7];
    const float* phi_e  = (const float*)d_in[8];
    const float* ke     = (const float*)d_in[9];
    const float* d2a    = (const float*)d_in[10];
    const float* e2m    = (const float*)d_in[11];
    float*       out    = (float*)d_out;

    const int       n_atoms = in_sizes[0];
    const long long n_pairs = in_sizes[1];

    float2* zza = (float2*)d_ws;                     // 8 B * n_atoms scratch

    const int g1 = (n_atoms + ZBL_BLOCK - 1) / ZBL_BLOCK;
    zbl_prep<<<g1, ZBL_BLOCK, 0, stream>>>(Z, a_exp, zza, out, n_atoms);

    const long long ntiles = n_pairs / ZBL_TILE;     // full tiles -> async path
    if (ntiles > 0) {
        int grid = (int)(ntiles < ZBL_MAIN_GRID ? ntiles : ZBL_MAIN_GRID);
        zbl_pairs_main<<<grid, ZBL_BLOCK, 0, stream>>>(
            dist, cut, idx_i, idx_j, zza,
            a_coef, phi_c, phi_e, ke, d2a, e2m, out, (int)ntiles);
    }
    const long long tail_base = ntiles * ZBL_TILE;
    if (tail_base < n_pairs) {                       // < ZBL_TILE tail pairs
        zbl_pairs_tail<<<1, ZBL_BLOCK, 0, stream>>>(
            dist, cut, idx_i, idx_j, zza,
            a_coef, phi_c, phi_e, ke, d2a, e2m, out, tail_base, n_pairs);
    }
}